// MAC_11776800325638
// MI455X (gfx1250) — compile-verified
//
#include <hip/hip_runtime.h>
#include <math.h>
#include <stdint.h>

// MinkowskiGlobalMaxPooling: segmented max over sorted batch indices.
// feats: [N, 64] f32, batch_idx: [N] i32 (sorted), out: [B, 64] f32.
// Memory-bound: 1 GiB stream @ 23.3 TB/s ~= 46 us floor. No matmul structure
// => no WMMA; optimize bytes/instruction: b128 loads, minimal per-row VALU,
// native gfx1250 f32 max atomics, LDS cross-walker reduction.

#define FEAT_D 64

__global__ void seg_init(float* __restrict__ out, int n) {
    int i = blockIdx.x * blockDim.x + threadIdx.x;
    if (i < n) out[i] = -INFINITY;
}

// Native CDNA5 float atomic max (no CAS loop). Device scope so RMWs from all
// WGPs resolve coherently at L2.
__device__ __forceinline__ void global_atomic_fmax(float* addr, float v) {
    asm volatile("global_atomic_max_num_f32 %0, %1, off scope:SCOPE_DEV"
                 :
                 : "v"((unsigned long long)(uintptr_t)addr), "v"(v)
                 : "memory");
}

__device__ __forceinline__ void fmax4(float4& a, const float4& b) {
    a.x = fmaxf(a.x, b.x);
    a.y = fmaxf(a.y, b.y);
    a.z = fmaxf(a.z, b.z);
    a.w = fmaxf(a.w, b.w);
}

__global__ void __launch_bounds__(256)
seg_max_kernel(const float* __restrict__ feats,
               const int* __restrict__ bidx,
               float* __restrict__ out,
               int nrows, int rows_per_block)
{
    const int tid = threadIdx.x;
    const int q   = tid & 15;   // feature quad: owns features [4q, 4q+4)
    const int w   = tid >> 4;   // walker id 0..15 (walks rows with stride 16)

    int row0 = blockIdx.x * rows_per_block;
    int rend = row0 + rows_per_block;
    if (rend > nrows) rend = nrows;
    if (row0 >= rend) return;

    const float4* __restrict__ feats4 = (const float4*)feats; // 16 float4/row

    // Block-uniform: does this chunk lie entirely inside one segment?
    const int b_first = bidx[row0];
    const int b_last  = bidx[rend - 1];

    __shared__ float4 red[16][16];   // [walker][quad], 4 KB

    float4 vmax = make_float4(-INFINITY, -INFINITY, -INFINITY, -INFINITY);

    if (b_first == b_last) {
        // ---- Fast path (~98% of blocks): pure streaming max ----
        #pragma unroll 4
        for (int r = row0 + w; r < rend; r += 16) {
            // L2 prefetch ~64 rows ahead, one lane per 64B line, clamped.
            if ((q & 3) == 0) {
                int rp = r + 64; rp = (rp < nrows) ? rp : (nrows - 1);
                __builtin_prefetch((const void*)(feats4 + rp * 16 + q), 0, 1);
            }
            fmax4(vmax, feats4[r * 16 + q]);   // global_load_b128 + 4x v_max
        }

        // Cross-walker LDS reduction: 16x fewer atomics.
        red[w][q] = vmax;
        __syncthreads();
        if (w == 0) {
            float4 m = red[0][q];
            #pragma unroll
            for (int i = 1; i < 16; ++i) fmax4(m, red[i][q]);
            float* o = out + b_first * FEAT_D + q * 4;
            global_atomic_fmax(o + 0, m.x);
            global_atomic_fmax(o + 1, m.y);
            global_atomic_fmax(o + 2, m.z);
            global_atomic_fmax(o + 3, m.w);
        }
        return;
    }

    // ---- Boundary path (rare: chunk spans >=2 segments) ----
    int cur_b = -1;
    for (int r = row0 + w; r < rend; r += 16) {
        int    b = bidx[r];               // broadcast within the walker
        float4 v = feats4[r * 16 + q];
        if (b != cur_b) {
            if (cur_b >= 0) {
                float* o = out + cur_b * FEAT_D + q * 4;
                global_atomic_fmax(o + 0, vmax.x);
                global_atomic_fmax(o + 1, vmax.y);
                global_atomic_fmax(o + 2, vmax.z);
                global_atomic_fmax(o + 3, vmax.w);
            }
            cur_b = b;
            vmax  = v;
        } else {
            fmax4(vmax, v);
        }
    }
    if (cur_b >= 0) {
        float* o = out + cur_b * FEAT_D + q * 4;
        global_atomic_fmax(o + 0, vmax.x);
        global_atomic_fmax(o + 1, vmax.y);
        global_atomic_fmax(o + 2, vmax.z);
        global_atomic_fmax(o + 3, vmax.w);
    }
}

extern "C" void kernel_launch(void* const* d_in, const int* in_sizes, int n_in,
                              void* d_out, int out_size, void* d_ws, size_t ws_size,
                              hipStream_t stream) {
    const float* feats = (const float*)d_in[0];
    const int*   bidx  = (const int*)d_in[1];
    float*       out   = (float*)d_out;

    const int nrows = in_sizes[1];          // N (batch_idx element count)

    // Re-initialize output to -inf every call (harness poisons buffers).
    int init_blocks = (out_size + 255) / 256;
    seg_init<<<init_blocks, 256, 0, stream>>>(out, out_size);

    // 1024 contiguous rows per block: with ~65K-row average segments, almost
    // every block is single-segment -> fast streaming path.
    const int rpb  = 1024;
    const int grid = (nrows + rpb - 1) / rpb;
    seg_max_kernel<<<grid, 256, 0, stream>>>(feats, bidx, out, nrows, rpb);
}